// TRBlock_25520695673111
// MI455X (gfx1250) — compile-verified
//
#include <hip/hip_runtime.h>
#include <hip/hip_bf16.h>

// ---------------------------------------------------------------------------
// Sparse voxel transformer block for MI455X (gfx1250, wave32).
// Strategy: invert the per-offset scatter (out indices unique per k) into a
// gather map inv[k][n], pad x/q/v with a zero dummy row N, then:
//   q-conv  : 27 gathered (16x128)x(128x16) tiles via V_WMMA_F32_16X16X4_F32
//   v-gemm  : dense 65536x128x128 via V_WMMA_F32_16X16X4_F32
//   attn/out: fused wave-per-row logits + softmax(27) + weighted v-accum
// Weights are pre-transposed so both A and B fragments load as single b64.
// ---------------------------------------------------------------------------

#define N_VOX  65536
#define PLANES 128
#define VD     16
#define KOFF   27

typedef float v2f __attribute__((ext_vector_type(2)));
typedef float v8f __attribute__((ext_vector_type(8)));

// ---- workspace layout (bytes) ---------------------------------------------
static const size_t XP_OFF  = 0;                                            // (N+1)*128 f32
static const size_t NZ_OFF  = XP_OFF + (size_t)(N_VOX + 1) * PLANES * 4;    // (N+1) f32
static const size_t INV_OFF = NZ_OFF + 262400;                              // 27*N i32
static const size_t QP_OFF  = INV_OFF + (size_t)KOFF * N_VOX * 4;           // (N+1)*16 f32
static const size_t VP_OFF  = QP_OFF + 4194560;                             // (N+1)*128 f32
static const size_t WQT_OFF = VP_OFF + (size_t)(N_VOX + 1) * PLANES * 4;    // 27*16*128 f32
static const size_t WVT_OFF = WQT_OFF + (size_t)KOFF * VD * PLANES * 4;     // 128*128 f32
// total ~78.9 MB

// ---------------------------------------------------------------------------
__global__ void __launch_bounds__(256)
k_prep(const float* __restrict__ x, float* __restrict__ xp, float* __restrict__ nz) {
    int wave = (blockIdx.x * blockDim.x + threadIdx.x) >> 5;
    int lane = threadIdx.x & 31;
    int nw   = (gridDim.x * blockDim.x) >> 5;
    for (int r = wave; r <= N_VOX; r += nw) {
        if (r == N_VOX) {
            float4 z = make_float4(0.f, 0.f, 0.f, 0.f);
            ((float4*)(xp + (size_t)r * PLANES))[lane] = z;
            if (lane == 0) nz[r] = 0.f;
        } else {
            float4 v = ((const float4*)(x + (size_t)r * PLANES))[lane];
            ((float4*)(xp + (size_t)r * PLANES))[lane] = v;
            int nzl = (v.x != 0.f) | (v.y != 0.f) | (v.z != 0.f) | (v.w != 0.f);
            int any = __any(nzl);
            if (lane == 0) nz[r] = any ? 1.f : 0.f;
        }
    }
}

// transpose weights: WqT[k][vd][p] = Wq[k][p][vd];  WvT[o][i] = Wv[i][o]
__global__ void __launch_bounds__(256)
k_wtrans(const float* __restrict__ Wq, const float* __restrict__ Wv,
         float* __restrict__ WqT, float* __restrict__ WvT) {
    int t = blockIdx.x * blockDim.x + threadIdx.x;
    if (t < KOFF * PLANES * VD) {
        int k  = t / (PLANES * VD);
        int r  = t - k * (PLANES * VD);
        int vd = r / PLANES;
        int p  = r - vd * PLANES;
        WqT[t] = Wq[k * PLANES * VD + p * VD + vd];
    }
    if (t < PLANES * PLANES) {
        int o = t / PLANES;
        int i = t - o * PLANES;
        WvT[t] = Wv[i * PLANES + o];
    }
}

__global__ void __launch_bounds__(256)
k_init_inv(int* __restrict__ inv, float* __restrict__ qp, float* __restrict__ vp) {
    int t = blockIdx.x * blockDim.x + threadIdx.x;
    if (t < KOFF * N_VOX) inv[t] = N_VOX;
    if (t < VD)     qp[(size_t)N_VOX * VD + t]     = 0.f;
    if (t < PLANES) vp[(size_t)N_VOX * PLANES + t] = 0.f;
}

__global__ void __launch_bounds__(256)
k_scatter_inv(const int* __restrict__ nin, const int* __restrict__ nout,
              int* __restrict__ inv) {
    int t = blockIdx.x * blockDim.x + threadIdx.x;
    if (t >= KOFF * N_VOX) return;
    int o = nout[t];
    if (o < N_VOX) inv[(t - (t & (N_VOX - 1))) + o] = nin[t];  // k*N + o
}

// ---------------------------------------------------------------------------
// q-conv: one wave per 16-row output tile.  C(16x16,f32) accumulates
// sum_k  gather(xp, inv[k]) (16x128)  @  Wq[k] (128x16)  via WMMA f32 K=4.
__global__ void __launch_bounds__(256)
k_qconv(const float* __restrict__ xp, const int* __restrict__ inv,
        const float* __restrict__ WqT, const float* __restrict__ qg,
        const float* __restrict__ qb, float* __restrict__ qp) {
    int wid  = (blockIdx.x * blockDim.x + threadIdx.x) >> 5;  // tile id
    int lane = threadIdx.x & 31;
    int half = lane >> 4;
    int m    = lane & 15;
    int row0 = wid * 16;

    v8f c = {0.f, 0.f, 0.f, 0.f, 0.f, 0.f, 0.f, 0.f};
    for (int k = 0; k < KOFF; ++k) {
        int idx = inv[k * N_VOX + row0 + m];
        const float* arow = xp  + (size_t)idx * PLANES;
        const float* brow = WqT + (size_t)k * PLANES * VD + m * PLANES;
        for (int kk = 0; kk < PLANES; kk += 4) {
            int ka = kk + 2 * half;
            v2f a = *(const v2f*)(arow + ka);   // A[m][ka..ka+1]
            v2f b = *(const v2f*)(brow + ka);   // B[ka..ka+1][m] (transposed)
            c = __builtin_amdgcn_wmma_f32_16x16x4_f32(false, a, false, b,
                                                      (short)0, c, false, false);
        }
    }
    float g = qg[m], be = qb[m];
#pragma unroll
    for (int j = 0; j < 8; ++j) {
        int r = row0 + j + 8 * half;
        float v = c[j] * g + be;
        qp[(size_t)r * VD + m] = v > 0.f ? v : 0.f;
    }
}

// ---------------------------------------------------------------------------
// v = relu((x @ Wv) * gamma + beta): one wave per 16-row tile, 8 N-tiles.
__global__ void __launch_bounds__(256)
k_vgemm(const float* __restrict__ xp, const float* __restrict__ WvT,
        const float* __restrict__ vg, const float* __restrict__ vb,
        float* __restrict__ vp) {
    int wid  = (blockIdx.x * blockDim.x + threadIdx.x) >> 5;
    int lane = threadIdx.x & 31;
    int half = lane >> 4;
    int m    = lane & 15;
    int row0 = wid * 16;

    v8f acc[8];
#pragma unroll
    for (int t = 0; t < 8; ++t) acc[t] = (v8f){0.f,0.f,0.f,0.f,0.f,0.f,0.f,0.f};

    const float* arow = xp + (size_t)(row0 + m) * PLANES;
    for (int kk = 0; kk < PLANES; kk += 4) {
        int ka = kk + 2 * half;
        v2f a = *(const v2f*)(arow + ka);
#pragma unroll
        for (int t = 0; t < 8; ++t) {
            // B[ka..ka+1][t*16+m] from WvT[out][in] layout -> contiguous b64
            v2f b = *(const v2f*)(WvT + (size_t)(t * 16 + m) * PLANES + ka);
            acc[t] = __builtin_amdgcn_wmma_f32_16x16x4_f32(false, a, false, b,
                                                           (short)0, acc[t], false, false);
        }
    }
#pragma unroll
    for (int t = 0; t < 8; ++t) {
        float g = vg[t * 16 + m], be = vb[t * 16 + m];
#pragma unroll
        for (int j = 0; j < 8; ++j) {
            int r = row0 + j + 8 * half;
            float v = acc[t][j] * g + be;
            vp[(size_t)r * PLANES + t * 16 + m] = v > 0.f ? v : 0.f;
        }
    }
}

// ---------------------------------------------------------------------------
// Fused logits + softmax(27) + weighted v accumulation + BN/ReLU + residual.
// One wave per output row n.  Lane keeps logits for vd = lane&15 in regs.
__global__ void __launch_bounds__(256)
k_attn_out(const float* __restrict__ x, const float* __restrict__ qp,
           const float* __restrict__ vp, const float* __restrict__ nz,
           const int* __restrict__ inv, const float* __restrict__ Wqk,
           const float* __restrict__ bqk, const float* __restrict__ og,
           const float* __restrict__ ob, float* __restrict__ out) {
    __shared__ float sWqk[VD * VD];
    __shared__ float sRel[8][VD];
    int tid = threadIdx.x;
    if (tid < VD * VD) sWqk[tid] = Wqk[tid];
    __syncthreads();

    int wv   = tid >> 5;
    int lane = tid & 31;
    int m    = lane & 15;
    int n    = blockIdx.x * 8 + wv;

    float wrow[VD];
#pragma unroll
    for (int j = 0; j < VD; ++j) wrow[j] = sWqk[m * VD + j];
    float bq = bqk[m];
    float qn = qp[(size_t)n * VD + m];

    float lg[KOFF];
#pragma unroll
    for (int k = 0; k < KOFF; ++k) {
        int   i   = inv[k * N_VOX + n];
        float msk = nz[i];
        float rel = qp[(size_t)i * VD + m] - qn * msk;
        if (lane < 16) sRel[wv][m] = rel;
        __syncthreads();
        float acc = bq;
#pragma unroll
        for (int jj = 0; jj < 4; ++jj) {
            float4 r4 = ((const float4*)(&sRel[wv][0]))[jj];
            acc += wrow[4 * jj]     * r4.x + wrow[4 * jj + 1] * r4.y
                 + wrow[4 * jj + 2] * r4.z + wrow[4 * jj + 3] * r4.w;
        }
        __syncthreads();
        lg[k] = acc * msk;
    }

    // softmax over the 27 offsets (masked entries contribute logit 0, as ref)
    float mx = lg[0];
#pragma unroll
    for (int k = 1; k < KOFF; ++k) mx = fmaxf(mx, lg[k]);
    float ssum = 0.f;
#pragma unroll
    for (int k = 0; k < KOFF; ++k) { lg[k] = __expf(lg[k] - mx); ssum += lg[k]; }
    float is = 1.f / ssum;

    // weighted accumulation: 4 channels per lane, attn[p/8] via shuffle
    float4 o4 = make_float4(0.f, 0.f, 0.f, 0.f);
#pragma unroll
    for (int k = 0; k < KOFF; ++k) {
        float a  = lg[k] * is;
        float av = __shfl(a, lane >> 1, 32);   // attn for vd = (lane*4)/8
        int   i  = inv[k * N_VOX + n];
        float w  = av * nz[i];
        float4 vv = ((const float4*)(vp + (size_t)i * PLANES))[lane];
        o4.x += vv.x * w; o4.y += vv.y * w; o4.z += vv.z * w; o4.w += vv.w * w;
    }

    float4 g4 = ((const float4*)og)[lane];
    float4 b4 = ((const float4*)ob)[lane];
    float4 x4 = ((const float4*)(x + (size_t)n * PLANES))[lane];
    float4 r;
    r.x = fmaxf(o4.x * g4.x + b4.x, 0.f) + x4.x;
    r.y = fmaxf(o4.y * g4.y + b4.y, 0.f) + x4.y;
    r.z = fmaxf(o4.z * g4.z + b4.z, 0.f) + x4.z;
    r.w = fmaxf(o4.w * g4.w + b4.w, 0.f) + x4.w;
    ((float4*)(out + (size_t)n * PLANES))[lane] = r;
}

// ---------------------------------------------------------------------------
extern "C" void kernel_launch(void* const* d_in, const int* in_sizes, int n_in,
                              void* d_out, int out_size, void* d_ws, size_t ws_size,
                              hipStream_t stream) {
    const float* x   = (const float*)d_in[0];
    const float* Wq  = (const float*)d_in[1];
    const float* Wv  = (const float*)d_in[2];
    const float* Wqk = (const float*)d_in[3];
    const float* bqk = (const float*)d_in[4];
    const float* qg  = (const float*)d_in[5];
    const float* qb  = (const float*)d_in[6];
    const float* vg  = (const float*)d_in[7];
    const float* vb  = (const float*)d_in[8];
    const float* og  = (const float*)d_in[9];
    const float* ob  = (const float*)d_in[10];
    const int*   nin = (const int*)d_in[11];
    const int*   nout= (const int*)d_in[12];

    char*  ws  = (char*)d_ws;
    float* xp  = (float*)(ws + XP_OFF);
    float* nz  = (float*)(ws + NZ_OFF);
    int*   inv = (int*)  (ws + INV_OFF);
    float* qp  = (float*)(ws + QP_OFF);
    float* vp  = (float*)(ws + VP_OFF);
    float* WqT = (float*)(ws + WQT_OFF);
    float* WvT = (float*)(ws + WVT_OFF);

    k_prep<<<512, 256, 0, stream>>>(x, xp, nz);
    k_wtrans<<<(KOFF * PLANES * VD + 255) / 256, 256, 0, stream>>>(Wq, Wv, WqT, WvT);

    int tot = KOFF * N_VOX;
    k_init_inv<<<(tot + 255) / 256, 256, 0, stream>>>(inv, qp, vp);
    k_scatter_inv<<<(tot + 255) / 256, 256, 0, stream>>>(nin, nout, inv);

    // N/16 = 4096 tiles, 8 waves (256 threads) per block
    k_qconv<<<N_VOX / 16 / 8, 256, 0, stream>>>(xp, inv, WqT, qg, qb, qp);
    k_vgemm<<<N_VOX / 16 / 8, 256, 0, stream>>>(xp, WvT, vg, vb, vp);

    k_attn_out<<<N_VOX / 8, 256, 0, stream>>>(x, qp, vp, nz, inv, Wqk, bqk,
                                              og, ob, (float*)d_out);
}